// CBAM_57810259804199
// MI455X (gfx1250) — compile-verified
//
#include <hip/hip_runtime.h>
#include <hip/hip_bf16.h>
#include <math.h>

#define ALPHA 0.02f
// Problem shape (fixed by reference): B=32, C=512, H=W=64, Cr=32
#define B_  32
#define C_  512
#define HW_ 4096   // 64*64
#define NCHUNK 16  // HW_ / 256
#define CB_ 64     // channels per LDS tile
#define TS_ 257    // padded LDS tile row stride (floats) -> conflict-free rows & cols

typedef float v2f __attribute__((ext_vector_type(2)));
typedef float v4f __attribute__((ext_vector_type(4)));
typedef float v8f __attribute__((ext_vector_type(8)));

#define NEGINF_ (-3.402823466e38f)

// ---------------------------------------------------------------------------
// Kernel 1: fused stats pass (single non-temporal read of x).
// grid = (16 hw-chunks, 32 batches), block = 256 threads (8 waves).
// Per 64x256 tile staged in LDS (stride 257: bank-conflict-free):
//   - spatial sum/max accumulate in REGISTERS during the load (each thread's
//     16 float4 loads cover the same 4 hw columns across 16 of the 64 rows, stride 4)
//   - channel sum/max read each element once row-wise from LDS (banks
//     (row+i) mod 64 across a wave -> conflict-free), 4-way partial combine
// No float atomics, fixed reduction order -> deterministic.
// ---------------------------------------------------------------------------
__global__ void __launch_bounds__(256)
cbam_stats_kernel(const float* __restrict__ x,
                  float* __restrict__ psum, float* __restrict__ pmax,
                  float* __restrict__ smean, float* __restrict__ smax)
{
    __shared__ float tile[CB_ * TS_];      // 65792 B
    __shared__ float prt_s[CB_][4];
    __shared__ float prt_m[CB_][4];
    __shared__ float sp_s[64][4][4];       // [colgroup][rowgroup][component]
    __shared__ float sp_m[64][4][4];

    const int chunk = blockIdx.x;          // 0..15
    const int b     = blockIdx.y;          // 0..31
    const int tid   = threadIdx.x;         // 0..255
    const int hw0   = chunk << 8;
    const float* xb = x + ((size_t)b << 21);   // b * C_ * HW_

    const int colg = tid & 63;             // column group: hw cols 4*colg..+3
    const int rgrp = tid >> 6;             // 0..3 (row offset within groups of 4)
    const int col4 = colg << 2;

    v4f ss4 = {0.f, 0.f, 0.f, 0.f};
    v4f sm4 = {NEGINF_, NEGINF_, NEGINF_, NEGINF_};

    for (int cb0 = 0; cb0 < C_; cb0 += CB_) {
        // ---- load tile (coalesced 512B/wave), spatial reg-accumulate, LDS store ----
        #pragma unroll
        for (int k = 0; k < 16; ++k) {
            const int row = (k << 2) + rgrp;                 // 0..63
            const v4f v = __builtin_nontemporal_load(
                reinterpret_cast<const v4f*>(xb + (((size_t)(cb0 + row)) << 12) + hw0 + col4));
            ss4 += v;
            sm4.x = fmaxf(sm4.x, v.x); sm4.y = fmaxf(sm4.y, v.y);
            sm4.z = fmaxf(sm4.z, v.z); sm4.w = fmaxf(sm4.w, v.w);
            float* t = &tile[row * TS_ + col4];
            t[0] = v.x; t[1] = v.y; t[2] = v.z; t[3] = v.w;
        }
        __syncthreads();

        // ---- channel partials: thread (cr, cs) reduces 64 row elements ----
        {
            const int cr = tid & 63;
            const int cs = tid >> 6;
            const float* rowp = &tile[cr * TS_ + (cs << 6)];
            float s = 0.f, m = NEGINF_;
            #pragma unroll
            for (int i = 0; i < 64; ++i) { const float v = rowp[i]; s += v; m = fmaxf(m, v); }
            prt_s[cr][cs] = s;
            prt_m[cr][cs] = m;
        }
        __syncthreads();

        if (tid < CB_) {
            const float s = prt_s[tid][0] + prt_s[tid][1] + prt_s[tid][2] + prt_s[tid][3];
            const float m = fmaxf(fmaxf(prt_m[tid][0], prt_m[tid][1]),
                                  fmaxf(prt_m[tid][2], prt_m[tid][3]));
            const int c = cb0 + tid;
            psum[(((b << 9) + c) << 4) + chunk] = s;
            pmax[(((b << 9) + c) << 4) + chunk] = m;
        }
        // next iteration's tile stores are fenced by its first __syncthreads;
        // prt reads above complete before any thread passes that barrier.
    }

    // ---- final spatial combine (once per block) ----
    sp_s[colg][rgrp][0] = ss4.x; sp_s[colg][rgrp][1] = ss4.y;
    sp_s[colg][rgrp][2] = ss4.z; sp_s[colg][rgrp][3] = ss4.w;
    sp_m[colg][rgrp][0] = sm4.x; sp_m[colg][rgrp][1] = sm4.y;
    sp_m[colg][rgrp][2] = sm4.z; sp_m[colg][rgrp][3] = sm4.w;
    __syncthreads();
    {
        const int p = tid;                 // hw position within chunk
        const int g = p >> 2, j = p & 3;
        float s = 0.f, m = NEGINF_;
        #pragma unroll
        for (int q = 0; q < 4; ++q) {
            s += sp_s[g][q][j];
            m = fmaxf(m, sp_m[g][q][j]);
        }
        smean[(b << 12) + hw0 + p] = s * (1.f / 512.f);
        smax [(b << 12) + hw0 + p] = m;
    }
}

// ---------------------------------------------------------------------------
// Kernel 2: reduce the 16 chunk-partials per (b,c) -> avg / max vectors.
// ---------------------------------------------------------------------------
__global__ void cbam_chanreduce_kernel(const float* __restrict__ psum,
                                       const float* __restrict__ pmax,
                                       float* __restrict__ avgv,
                                       float* __restrict__ maxv)
{
    const int i = blockIdx.x * blockDim.x + threadIdx.x;  // 0..16383
    float s = 0.f, m = NEGINF_;
    #pragma unroll
    for (int j = 0; j < NCHUNK; ++j) {
        s += psum[(i << 4) + j];
        m = fmaxf(m, pmax[(i << 4) + j]);
    }
    avgv[i] = s * (1.f / 4096.f);
    maxv[i] = m;
}

// ---------------------------------------------------------------------------
// Kernel 3: channel-attention MLP with V_WMMA_F32_16X16X4_F32.
// One block, 8 waves.
//   Phase 1: h1 = relu(V @ w1^T), V in {avg(32x512), max(32x512)} -> LDS 64x32
//            8 tiles of 16x16, one per wave, K=512 as 128 chained WMMAs.
//   Phase 2: mc = sigmoid( relu(h1_avg)@w2^T + relu(h1_max)@w2^T )
//            64 tiles (32x512 output), 8 per wave; avg/max contributions share
//            one WMMA accumulator chain (shared MLP weights).
// Fragment layouts per ISA 7.12.2 (A 16x4: lanes 0-15 hold K0/K1, lanes 16-31
// hold K2/K3; C/D: VGPR v <-> row v (lanes 0-15) / v+8 (lanes 16-31)).
// ---------------------------------------------------------------------------
__global__ void __launch_bounds__(256)
cbam_mlp_kernel(const float* __restrict__ avgv, const float* __restrict__ maxv,
                const float* __restrict__ w1,   const float* __restrict__ w2,
                float* __restrict__ mc)
{
    __shared__ float h1r[64 * 32];    // relu(h1): rows 0-31 avg path, 32-63 max path
    const int tid  = threadIdx.x;
    const int lane = tid & 31;
    const int wid  = tid >> 5;
    const int r    = lane & 15;
    const int koff = (lane < 16) ? 0 : 2;
    const int rowh = (lane < 16) ? 0 : 8;

    // ---- Phase 1 ----
    {
        const int matIdx = wid >> 2;        // 0=avg, 1=max
        const int mT     = (wid >> 1) & 1;  // M tile (32 rows -> 2 tiles)
        const int nT     = wid & 1;         // N tile (32 cols -> 2 tiles)
        const float* V   = matIdx ? maxv : avgv;
        const int m = mT * 16 + r;
        const int n = nT * 16 + r;
        v8f acc = {};
        for (int kb = 0; kb < 512; kb += 4) {
            v2f a, bf;
            a.x  = V[m * 512 + kb + koff];
            a.y  = V[m * 512 + kb + koff + 1];
            bf.x = w1[n * 512 + kb + koff];      // B[k][n] = w1[n][k]
            bf.y = w1[n * 512 + kb + koff + 1];
            acc = __builtin_amdgcn_wmma_f32_16x16x4_f32(
                false, a, false, bf, (short)0, acc, false, false);
        }
        const int rowbase = matIdx * 32 + mT * 16 + rowh;
        const int col     = nT * 16 + r;
        #pragma unroll
        for (int v = 0; v < 8; ++v)
            h1r[(rowbase + v) * 32 + col] = fmaxf(acc[v], 0.f);
    }
    __syncthreads();

    // ---- Phase 2 ----
    for (int t = 0; t < 8; ++t) {
        const int tile = wid * 8 + t;   // 0..63
        const int mT   = tile >> 5;     // 0..1  (32 output rows)
        const int nT   = tile & 31;     // 0..31 (512 output cols)
        const int n    = nT * 16 + r;
        v8f acc = {};
        #pragma unroll
        for (int p = 0; p < 2; ++p) {   // p=0: avg path, p=1: max path (chained)
            const int arow = p * 32 + mT * 16 + r;
            #pragma unroll
            for (int kb = 0; kb < 32; kb += 4) {
                v2f a, bf;
                a.x  = h1r[arow * 32 + kb + koff];
                a.y  = h1r[arow * 32 + kb + koff + 1];
                bf.x = w2[n * 32 + kb + koff];   // B[k][n] = w2[n][k]
                bf.y = w2[n * 32 + kb + koff + 1];
                acc = __builtin_amdgcn_wmma_f32_16x16x4_f32(
                    false, a, false, bf, (short)0, acc, false, false);
            }
        }
        const int rowbase = mT * 16 + rowh;  // batch index b
        const int col     = nT * 16 + r;     // channel index c
        #pragma unroll
        for (int v = 0; v < 8; ++v) {
            const float val = acc[v];
            mc[(rowbase + v) * 512 + col] = 1.f / (1.f + __expf(-val));
        }
    }
}

// ---------------------------------------------------------------------------
// Kernel 4: 7x7 spatial conv on [mean;max] maps + sigmoid -> ms [B,HW].
// ---------------------------------------------------------------------------
__global__ void __launch_bounds__(256)
cbam_sconv_kernel(const float* __restrict__ smean, const float* __restrict__ smax,
                  const float* __restrict__ wconv, float* __restrict__ ms)
{
    __shared__ float wk[98];
    const int tid = threadIdx.x;
    if (tid < 98) wk[tid] = wconv[tid];
    __syncthreads();

    const int b  = blockIdx.x >> 4;
    const int hw = ((blockIdx.x & 15) << 8) + tid;
    const int h  = hw >> 6, w = hw & 63;
    const float* pm = smean + (b << 12);
    const float* px = smax  + (b << 12);

    float acc = 0.f;
    #pragma unroll
    for (int i = 0; i < 7; ++i) {
        const int hh = h + i - 3;
        if (hh < 0 || hh >= 64) continue;
        #pragma unroll
        for (int j = 0; j < 7; ++j) {
            const int ww = w + j - 3;
            if (ww < 0 || ww >= 64) continue;
            const int off = (hh << 6) + ww;
            acc += wk[i * 7 + j] * pm[off] + wk[49 + i * 7 + j] * px[off];
        }
    }
    ms[(b << 12) + hw] = 1.f / (1.f + __expf(-acc));
}

// ---------------------------------------------------------------------------
// Kernel 5: gating pass, float4-vectorized, non-temporal x/out streaming.
// ---------------------------------------------------------------------------
__global__ void __launch_bounds__(256)
cbam_apply_kernel(const float* __restrict__ x, const float* __restrict__ mc,
                  const float* __restrict__ ms, float* __restrict__ out)
{
    const size_t i4   = (size_t)blockIdx.x * 256 + threadIdx.x; // 16M float4s
    const size_t base = i4 * 4;
    const int hw = (int)(base & 4095);
    const int c  = (int)((base >> 12) & 511);
    const int b  = (int)(base >> 21);

    const float gc = 1.f - ALPHA + ALPHA * mc[(b << 9) + c];
    const v4f xv = __builtin_nontemporal_load(reinterpret_cast<const v4f*>(x) + i4);
    const v4f mv = *reinterpret_cast<const v4f*>(ms + ((size_t)b << 12) + hw);
    v4f o;
    o.x = xv.x * gc * (1.f - ALPHA + ALPHA * mv.x);
    o.y = xv.y * gc * (1.f - ALPHA + ALPHA * mv.y);
    o.z = xv.z * gc * (1.f - ALPHA + ALPHA * mv.z);
    o.w = xv.w * gc * (1.f - ALPHA + ALPHA * mv.w);
    __builtin_nontemporal_store(o, reinterpret_cast<v4f*>(out) + i4);
}

// ---------------------------------------------------------------------------
extern "C" void kernel_launch(void* const* d_in, const int* in_sizes, int n_in,
                              void* d_out, int out_size, void* d_ws, size_t ws_size,
                              hipStream_t stream)
{
    const float* x     = (const float*)d_in[0];   // [32,512,64,64]
    const float* w1    = (const float*)d_in[1];   // [32,512]
    const float* w2    = (const float*)d_in[2];   // [512,32]
    const float* wconv = (const float*)d_in[3];   // [1,2,7,7] = 98 floats
    float* out = (float*)d_out;

    // Workspace layout (floats, all regions multiple of 4 elements)
    float* ws    = (float*)d_ws;
    float* psum  = ws;                    // 32*512*16 = 262144
    float* pmax  = psum  + (B_*C_*NCHUNK);// 262144
    float* avgv  = pmax  + (B_*C_*NCHUNK);// 16384
    float* maxv  = avgv  + (B_*C_);       // 16384
    float* smean = maxv  + (B_*C_);       // 131072
    float* smaxb = smean + (B_*HW_);      // 131072
    float* mc    = smaxb + (B_*HW_);      // 16384
    float* ms    = mc    + (B_*C_);       // 131072
    (void)in_sizes; (void)n_in; (void)out_size; (void)ws_size;

    // 1) fused stats: one pass over x (both channel and spatial pooling)
    cbam_stats_kernel<<<dim3(NCHUNK, B_), 256, 0, stream>>>(x, psum, pmax, smean, smaxb);
    // 2) chunk reduction -> pooled vectors
    cbam_chanreduce_kernel<<<(B_*C_)/256, 256, 0, stream>>>(psum, pmax, avgv, maxv);
    // 3) channel-attention MLP (WMMA f32 16x16x4), one workgroup
    cbam_mlp_kernel<<<1, 256, 0, stream>>>(avgv, maxv, w1, w2, mc);
    // 4) spatial 7x7 conv + sigmoid
    cbam_sconv_kernel<<<(B_*HW_)/256, 256, 0, stream>>>(smean, smaxb, wconv, ms);
    // 5) gating pass (read x again + write out)
    cbam_apply_kernel<<<(B_*C_*HW_)/(4*256), 256, 0, stream>>>(x, mc, ms, out);
}